// LearnableDetectionLoss_16973710754143
// MI455X (gfx1250) — compile-verified
//
#include <hip/hip_runtime.h>
#include <hip/hip_bf16.h>

#define B_   16
#define C_   80
#define H_   64
#define W_   64
#define K_   20
#define HW_  4096

// 10 * log2(e): sigmoid((v-thr)*10) = rcp(1 + exp2((thr-v)*10*log2e))
#define PEAK_TEMP_LOG2E_ 14.426950408889634f

typedef __attribute__((ext_vector_type(2))) float v2f;
typedef __attribute__((ext_vector_type(8))) float v8f;

// D = A(16x4 f32) x B(4x16 f32) + C(16x16 f32), full-FP32 matrix pipe.
// With B == all-ones, D[m][n] = rowsum(A[m]) for every n -> accumulating
// row-sum reduction in the WMMA unit.
static __device__ inline v8f wmma_rowsum_acc(v2f a, v2f ones, v8f c) {
    return __builtin_amdgcn_wmma_f32_16x16x4_f32(
        /*neg_a=*/false, a, /*neg_b=*/false, ones,
        /*c_mod=*/(short)0, c, /*reuse_a=*/false, /*reuse_b=*/false);
}

// sigmoid((v - thr) * 10) via hardware v_exp_f32 / v_rcp_f32 (1-ulp HW ops,
// ~1e-7 rel error vs the IEEE expf+fdiv sequence -> negligible for a loss).
static __device__ inline float peak_sigmoid(float v, float thr) {
    const float e = __builtin_amdgcn_exp2f((thr - v) * PEAK_TEMP_LOG2E_);
    return __builtin_amdgcn_rcpf(1.0f + e);
}

__global__ __launch_bounds__(256)
void loss_main_kernel(const float* __restrict__ cam,
                      const float* __restrict__ pboxes,
                      const float* __restrict__ gboxes,
                      const int*   __restrict__ glabels,
                      const float* __restrict__ thr_p,
                      float* __restrict__ partial)
{
    const int bk   = blockIdx.x;          // 0..319  = b*K + k
    const int b    = bk / K_;
    const int tid  = threadIdx.x;
    const int lane = tid & 31;
    const int wave = tid >> 5;

    const int   label = glabels[bk];
    const float gx1 = gboxes[bk * 4 + 0];
    const float gy1 = gboxes[bk * 4 + 1];
    const float gx2 = gboxes[bk * 4 + 2];
    const float gy2 = gboxes[bk * 4 + 3];
    const float thr = thr_p[0];

    const int i_min = (int)fmaxf(0.0f,      floorf(gy1 * (float)H_));
    const int i_max = (int)fminf(H_ - 1.0f, floorf(gy2 * (float)H_));
    const int j_min = (int)fmaxf(0.0f,      floorf(gx1 * (float)W_));
    const int j_max = (int)fminf(W_ - 1.0f, floorf(gx2 * (float)W_));

    const float* __restrict__ cb = cam + (size_t)(b * C_ + label) * HW_;

    const int m  = lane & 15;   // A-matrix row
    const int kh = lane >> 4;   // 0: K=0,1   1: K=2,3

    v8f acc_cam = {0.f,0.f,0.f,0.f,0.f,0.f,0.f,0.f};
    v8f acc_cin = {0.f,0.f,0.f,0.f,0.f,0.f,0.f,0.f};
    v8f acc_pk  = {0.f,0.f,0.f,0.f,0.f,0.f,0.f,0.f};
    v8f acc_pin = {0.f,0.f,0.f,0.f,0.f,0.f,0.f,0.f};
    const v2f ones2 = {1.0f, 1.0f};

    float bestv = -__builtin_inff();
    int   besti = 0x7FFFFFFF;

    // 8 waves/block, each wave owns image row (t*8 + wave) per iteration t.
    #pragma unroll
    for (int t = 0; t < 8; ++t) {
        const int row = t * 8 + wave;            // 0..63
        const int j0  = 4 * m + 2 * kh;          // column of first element
        const int q   = row * W_ + j0;           // flat pixel index (even -> 8B aligned)

        // speculative prefetch of next iteration's row (global_prefetch_b8);
        // overshoot past the tile on t==7 is silently dropped by hardware.
        __builtin_prefetch(cb + q + 8 * W_, 0, 1);

        const v2f val = *(const v2f*)(cb + q);
        const float v0 = val.x, v1 = val.y;

        const bool rin = (row >= i_min) & (row <= i_max);
        const float m0 = (rin & (j0     >= j_min) & (j0     <= j_max)) ? 1.0f : 0.0f;
        const float m1 = (rin & (j0 + 1 >= j_min) & (j0 + 1 <= j_max)) ? 1.0f : 0.0f;

        const float p0 = peak_sigmoid(v0, thr);
        const float p1 = peak_sigmoid(v1, thr);

        const v2f a_cam = {v0,        v1       };
        const v2f a_cin = {v0 * m0,   v1 * m1  };
        const v2f a_pk  = {p0,        p1       };
        const v2f a_pin = {p0 * m0,   p1 * m1  };

        acc_cam = wmma_rowsum_acc(a_cam, ones2, acc_cam);
        acc_cin = wmma_rowsum_acc(a_cin, ones2, acc_cin);
        acc_pk  = wmma_rowsum_acc(a_pk,  ones2, acc_pk);
        acc_pin = wmma_rowsum_acc(a_pin, ones2, acc_pin);

        // masked argmax (first-occurrence tie-break, like jnp.argmax)
        const float mv0 = (m0 > 0.0f) ? v0 : -__builtin_inff();
        const float mv1 = (m1 > 0.0f) ? v1 : -__builtin_inff();
        bool bet = (mv0 > bestv) || ((mv0 == bestv) && (q < besti));
        bestv = bet ? mv0 : bestv;  besti = bet ? q : besti;
        bet = (mv1 > bestv) || ((mv1 == bestv) && ((q + 1) < besti));
        bestv = bet ? mv1 : bestv;  besti = bet ? (q + 1) : besti;
    }

    // D layout: lanes 0-15 hold M=0..7 in v8f, lanes 16-31 hold M=8..15.
    // Columns are identical (B == ones), so lanes 0 and 16 carry the two halves.
    __shared__ float s_sums[4][16];
    __shared__ float s_val[256];
    __shared__ int   s_idx[256];

    if (lane == 0 || lane == 16) {
        float s0 = 0.f, s1 = 0.f, s2 = 0.f, s3 = 0.f;
        #pragma unroll
        for (int e = 0; e < 8; ++e) {
            s0 += acc_cam[e]; s1 += acc_cin[e]; s2 += acc_pk[e]; s3 += acc_pin[e];
        }
        const int slot = wave * 2 + kh;
        s_sums[0][slot] = s0; s_sums[1][slot] = s1;
        s_sums[2][slot] = s2; s_sums[3][slot] = s3;
    }
    s_val[tid] = bestv;
    s_idx[tid] = besti;
    __syncthreads();

    // block-wide argmax tree
    for (int off = 128; off > 0; off >>= 1) {
        if (tid < off) {
            const float ov = s_val[tid + off];
            const int   oi = s_idx[tid + off];
            const bool bet = (ov > s_val[tid]) || ((ov == s_val[tid]) && (oi < s_idx[tid]));
            if (bet) { s_val[tid] = ov; s_idx[tid] = oi; }
        }
        __syncthreads();
    }

    if (tid == 0) {
        float sum_cam = 0.f, sum_cin = 0.f, sum_pk = 0.f, sum_pin = 0.f;
        #pragma unroll
        for (int w = 0; w < 16; ++w) {
            sum_cam += s_sums[0][w]; sum_cin += s_sums[1][w];
            sum_pk  += s_sums[2][w]; sum_pin += s_sums[3][w];
        }

        const bool nonempty = (i_max >= i_min) && (j_max >= j_min);
        const float s_in  = nonempty ? (float)((i_max - i_min + 1) * (j_max - j_min + 1)) : 0.0f;
        const float s_out = (float)HW_ - s_in;

        const float v_in  = sum_cin / fmaxf(s_in, 1.0f);
        const float v_out = (sum_cam - sum_cin) / fmaxf(s_out, 1.0f);
        const float cam_term  = ((s_in  > 0.f) ? (1.0f - v_in) : 0.0f)
                              + ((s_out > 0.f) ? v_out : 0.0f);

        const float pv_in  = sum_pin / fmaxf(s_in, 1.0f);
        const float pv_out = (sum_pk - sum_pin) / fmaxf(s_out, 1.0f);
        const float peak_term = ((s_in  > 0.f) ? (1.0f - pv_in) : 0.0f)
                              + ((s_out > 0.f) ? pv_out : 0.0f);

        int idx = s_idx[0];
        if (!(s_val[0] > -__builtin_inff())) idx = 0;   // all -inf -> argmax = 0
        const int ii = idx >> 6, jj = idx & 63;

        const float* pp = pboxes + ((((size_t)(b * C_ + label)) * H_ + ii) * W_ + jj) * 4;
        const float px1 = pp[0], py1 = pp[1], px2 = pp[2], py2 = pp[3];

        const float l1 = fabsf(px1 - gx1) + fabsf(py1 - gy1)
                       + fabsf(px2 - gx2) + fabsf(py2 - gy2);

        const float ltx = fmaxf(px1, gx1), lty = fmaxf(py1, gy1);
        const float rbx = fminf(px2, gx2), rby = fminf(py2, gy2);
        const float iw = fmaxf(rbx - ltx, 0.0f), ih = fmaxf(rby - lty, 0.0f);
        const float inter = iw * ih;
        const float ap = (px2 - px1) * (py2 - py1);
        const float ag = (gx2 - gx1) * (gy2 - gy1);
        const float uni = ap + ag - inter;
        const float iou = inter / uni;
        const float cltx = fminf(px1, gx1), clty = fminf(py1, gy1);
        const float crbx = fmaxf(px2, gx2), crby = fmaxf(py2, gy2);
        const float cw = fmaxf(crbx - cltx, 0.0f), ch = fmaxf(crby - clty, 0.0f);
        const float ac = cw * ch;
        const float giou = iou - (ac - uni) / ac;

        // weights: L1=1.0 (mean over B*K*4), GIOU=2.0, CAM=0.5, PEAK=0.5
        const float part = (1.0f * (l1 * 0.25f)
                          + 2.0f * (1.0f - giou)
                          + 0.5f * cam_term
                          + 0.5f * peak_term) * (1.0f / (float)(B_ * K_));
        partial[bk] = part;
    }
}

__global__ __launch_bounds__(256)
void loss_reduce_kernel(const float* __restrict__ partial, float* __restrict__ out)
{
    __shared__ float s[256];
    const int tid = threadIdx.x;
    float v = partial[tid];                 // 256 of 320
    if (tid < B_ * K_ - 256) v += partial[tid + 256];
    s[tid] = v;
    __syncthreads();
    for (int off = 128; off > 0; off >>= 1) {
        if (tid < off) s[tid] += s[tid + off];
        __syncthreads();
    }
    if (tid == 0) out[0] = s[0];
}

extern "C" void kernel_launch(void* const* d_in, const int* in_sizes, int n_in,
                              void* d_out, int out_size, void* d_ws, size_t ws_size,
                              hipStream_t stream) {
    const float* cam    = (const float*)d_in[0];
    const float* pboxes = (const float*)d_in[1];
    const float* gboxes = (const float*)d_in[2];
    const int*   glabel = (const int*)d_in[3];
    const float* thr    = (const float*)d_in[4];
    float* ws  = (float*)d_ws;
    float* out = (float*)d_out;

    loss_main_kernel<<<B_ * K_, 256, 0, stream>>>(cam, pboxes, gboxes, glabel, thr, ws);
    loss_reduce_kernel<<<1, 256, 0, stream>>>(ws, out);
}